// GCN_BL_56899726737497
// MI455X (gfx1250) — compile-verified
//
#include <hip/hip_runtime.h>

typedef __attribute__((ext_vector_type(2))) float v2f;
typedef __attribute__((ext_vector_type(8))) float v8f;

#define D 128
#define TILE_ROWS 128
#define SLOPE 0.01f

// ---------------- degree / norm ----------------
__global__ void k_degrees(const int* __restrict__ src, const int* __restrict__ dst,
                          float* __restrict__ outdeg, float* __restrict__ indeg, int E) {
  int e = blockIdx.x * blockDim.x + threadIdx.x;
  if (e < E) {
    atomicAdd(&outdeg[src[e]], 1.0f);
    atomicAdd(&indeg[dst[e]], 1.0f);
  }
}

__global__ void k_norms(const float* __restrict__ outdeg, const float* __restrict__ indeg,
                        float* __restrict__ nout, float* __restrict__ nin, int n) {
  int i = blockIdx.x * blockDim.x + threadIdx.x;
  if (i < n) {
    nout[i] = rsqrtf(fmaxf(outdeg[i], 1.0f));
    nin[i]  = rsqrtf(fmaxf(indeg[i], 1.0f));
  }
}

// ---------------- fold W2 @ agg_w into a single 128-vector ----------------
__global__ void k_w2v(const float* __restrict__ W2, const float* __restrict__ aggw,
                      const float* __restrict__ b2, const float* __restrict__ aggb,
                      float* __restrict__ w2v, float* __restrict__ cbias) {
  int k = threadIdx.x;  // 0..127
  float s = 0.f;
  for (int j = 0; j < D; ++j) s += W2[k * D + j] * aggw[j];
  w2v[k] = s;
  if (k == 0) {
    float c = 0.f;
    for (int j = 0; j < D; ++j) c += b2[j] * aggw[j];
    cbias[0] = c + aggb[0];
  }
}

// ---------------- edge scatter: one wave per edge, float4 per lane ----------------
__global__ void __launch_bounds__(256) k_scatter(const float* __restrict__ feat,
    const int* __restrict__ src, const int* __restrict__ dst,
    const float* __restrict__ scale, float* __restrict__ agg, int E, int use_scale) {
  int wid  = (blockIdx.x * blockDim.x + threadIdx.x) >> 5;
  int lane = threadIdx.x & 31;
  if (wid >= E) return;
  int s = src[wid], d = dst[wid];
  float ns = use_scale ? scale[s] : 1.0f;
  float4 v = ((const float4*)(feat + (size_t)s * D))[lane];
  float* a = agg + (size_t)d * D + lane * 4;
  atomicAdd(a + 0, v.x * ns);
  atomicAdd(a + 1, v.y * ns);
  atomicAdd(a + 2, v.z * ns);
  atomicAdd(a + 3, v.w * ns);
}

// ---------------- WMMA GEMM: out = postscale(leaky((nin.*A) @ W + b)) ----------------
// LDS holds W pre-paired along K: Wp[kp*D + col] = {W[2kp][col], W[2kp+1][col]},
// so every B fragment is one contiguous ds_load_b64 (no repacking movs).
__global__ void __launch_bounds__(256) k_gcn_gemm_wmma(
    const float* __restrict__ A,     // [n, D] raw aggregate
    const float* __restrict__ nin,   // per-row pre-scale
    const float* __restrict__ W,     // [D, D] row-major (K x N)
    const float* __restrict__ bias,  // [D]
    const float* __restrict__ nout,  // per-row post-scale (next layer's norm_out)
    float* __restrict__ out, int n) {
  __shared__ v2f Wp[(D / 2) * D];  // 64 KB of the 320 KB WGP LDS
  for (int i = threadIdx.x; i < (D / 2) * D; i += 256) {
    int kp = i >> 7, col = i & (D - 1);
    v2f w;
    w.x = W[(2 * kp) * D + col];
    w.y = W[(2 * kp + 1) * D + col];
    Wp[i] = w;
  }
  __syncthreads();

  int lane = threadIdx.x & 31;
  int wv   = threadIdx.x >> 5;   // 8 waves, each owns a 16-row stripe
  int half = lane >> 4;          // 0: K=k..k+1, 1: K=k+2..k+3
  int l16  = lane & 15;
  int row0 = blockIdx.x * TILE_ROWS + wv * 16;
  int arow = min(row0 + l16, n - 1);
  float rs = nin[arow];
  const float* Arow = A + (size_t)arow * D;
  int koff = half * 2;
  const v2f* Wlane = Wp + half * D + l16;  // + (k/2)*D + t*16 per step/tile

  v8f acc[8] = {};

  for (int k = 0; k < D; k += 4) {
    v2f a;
    a.x = Arow[k + koff] * rs;
    a.y = Arow[k + koff + 1] * rs;
    const v2f* Wk = Wlane + (k >> 1) * D;
#pragma unroll
    for (int t = 0; t < 8; ++t) {  // 8 column tiles, A fragment reused
      v2f b = Wk[t * 16];
      acc[t] = __builtin_amdgcn_wmma_f32_16x16x4_f32(
          false, a, false, b, (short)0, acc[t], false, false);
    }
  }

#pragma unroll
  for (int t = 0; t < 8; ++t) {
    int col = t * 16 + l16;
    float bv = bias[col];
#pragma unroll
    for (int r = 0; r < 8; ++r) {
      int row = row0 + r + half * 8;
      if (row < n) {
        float v = acc[t][r] + bv;
        v = (v >= 0.f) ? v : SLOPE * v;
        out[(size_t)row * D + col] = v * nout[row];
      }
    }
  }
}

// ---------------- s[n] = nin[n] * dot(agg2[n,:], w2v) + cbias ----------------
__global__ void __launch_bounds__(256) k_rowdot(const float* __restrict__ agg,
    const float* __restrict__ nin, const float* __restrict__ w2v,
    const float* __restrict__ cbias, float* __restrict__ s, int n) {
  int wid  = (blockIdx.x * blockDim.x + threadIdx.x) >> 5;
  int lane = threadIdx.x & 31;
  if (wid >= n) return;
  float4 a = ((const float4*)(agg + (size_t)wid * D))[lane];
  float4 w = ((const float4*)w2v)[lane];
  float p = a.x * w.x + a.y * w.y + a.z * w.z + a.w * w.w;
#pragma unroll
  for (int off = 16; off > 0; off >>= 1) p += __shfl_xor(p, off, 32);
  if (lane == 0) s[wid] = p * nin[wid] + cbias[0];
}

// ---------------- z1 = s @ d1_w + d1_b ----------------
__global__ void k_z1_init(const float* __restrict__ d1b, float* __restrict__ z1) {
  int j = threadIdx.x;
  if (j < 100) z1[j] = d1b[j];
}

__global__ void __launch_bounds__(128) k_z1_acc(const float* __restrict__ s,
    const float* __restrict__ d1w, float* __restrict__ z1, int n) {
  int j  = threadIdx.x;              // column, active < 100 (coalesced over 100 cols)
  int r0 = blockIdx.x * 512;
  int r1 = min(r0 + 512, n);
  float acc = 0.f;
  for (int r = r0; r < r1; ++r) {
    float sv = s[r];
    if (j < 100) {
      __builtin_prefetch(&d1w[(size_t)(r + 16) * 100 + j], 0, 1);  // global_prefetch_b8
      acc += sv * d1w[(size_t)r * 100 + j];
    }
  }
  if (j < 100) atomicAdd(&z1[j], acc);
}

// ---------------- tiny head: leaky(z1@d2w+b) @ d3w + b ----------------
__global__ void k_head(const float* __restrict__ z1, const float* __restrict__ d2w,
                       const float* __restrict__ d2b, const float* __restrict__ d3w,
                       const float* __restrict__ d3b, float* __restrict__ out) {
  __shared__ float z2[20];
  int t = threadIdx.x;
  if (t < 20) {
    float a = d2b[t];
    for (int k = 0; k < 100; ++k) a += z1[k] * d2w[k * 20 + t];
    z2[t] = (a >= 0.f) ? a : SLOPE * a;
  }
  __syncthreads();
  if (t < 10) {
    float a = d3b[t];
    for (int k = 0; k < 20; ++k) a += z2[k] * d3w[k * 10 + t];
    out[t] = a;
  }
}

extern "C" void kernel_launch(void* const* d_in, const int* in_sizes, int n_in,
                              void* d_out, int out_size, void* d_ws, size_t ws_size,
                              hipStream_t stream) {
  const float* x    = (const float*)d_in[0];
  const int*   src  = (const int*)d_in[1];
  const int*   dst  = (const int*)d_in[2];
  const float* W1   = (const float*)d_in[3];
  const float* b1   = (const float*)d_in[4];
  const float* W2   = (const float*)d_in[5];
  const float* b2   = (const float*)d_in[6];
  const float* aggw = (const float*)d_in[7];
  const float* aggb = (const float*)d_in[8];
  const float* d1w  = (const float*)d_in[9];
  const float* d1b  = (const float*)d_in[10];
  const float* d2w  = (const float*)d_in[11];
  const float* d2b  = (const float*)d_in[12];
  const float* d3w  = (const float*)d_in[13];
  const float* d3b  = (const float*)d_in[14];

  const int N = in_sizes[0] / D;
  const int E = in_sizes[1];

  char* p = (char*)d_ws;
  auto carve = [&](size_t bytes) {
    char* r = p;
    p += (bytes + 255) & ~(size_t)255;
    return r;
  };
  float* deg  = (float*)carve((size_t)2 * N * 4);   // outdeg | indeg
  float* nout = (float*)carve((size_t)N * 4);
  float* nin  = (float*)carve((size_t)N * 4);
  float* agg  = (float*)carve((size_t)N * D * 4);   // reused for both layers
  float* h1s  = (float*)carve((size_t)N * D * 4);
  float* svec = (float*)carve((size_t)N * 4);
  float* w2v  = (float*)carve(D * 4);
  float* cb   = (float*)carve(4);
  float* z1   = (float*)carve(100 * 4);
  (void)ws_size; (void)n_in; (void)out_size;

  hipMemsetAsync(deg, 0, (size_t)2 * N * 4, stream);
  k_degrees<<<(E + 255) / 256, 256, 0, stream>>>(src, dst, deg, deg + N, E);
  k_norms<<<(N + 255) / 256, 256, 0, stream>>>(deg, deg + N, nout, nin, N);
  k_w2v<<<1, D, 0, stream>>>(W2, aggw, b2, aggb, w2v, cb);

  // conv1: scatter(x * nout) -> agg; (nin.*agg)@W1+b1 -> leaky -> *nout -> h1s
  hipMemsetAsync(agg, 0, (size_t)N * D * 4, stream);
  k_scatter<<<(E + 7) / 8, 256, 0, stream>>>(x, src, dst, nout, agg, E, 1);
  k_gcn_gemm_wmma<<<(N + TILE_ROWS - 1) / TILE_ROWS, 256, 0, stream>>>(
      agg, nin, W1, b1, nout, h1s, N);

  // conv2 (collapsed with agg_w): scatter(h1s) -> agg; s = nin .* (agg @ w2v) + cb
  hipMemsetAsync(agg, 0, (size_t)N * D * 4, stream);
  k_scatter<<<(E + 7) / 8, 256, 0, stream>>>(h1s, src, dst, nullptr, agg, E, 0);
  k_rowdot<<<(N + 7) / 8, 256, 0, stream>>>(agg, nin, w2v, cb, svec, N);

  // dense head
  k_z1_init<<<1, 128, 0, stream>>>(d1b, z1);
  k_z1_acc<<<(N + 511) / 512, 128, 0, stream>>>(svec, d1w, z1, N);
  k_head<<<1, 32, 0, stream>>>(z1, d2w, d2b, d3w, d3b, (float*)d_out);
}